// SobelLayer_12833362280675
// MI455X (gfx1250) — compile-verified
//
#include <hip/hip_runtime.h>

#define IMG_W 1024
#define IMG_H 1024
#define TILE_W 256
#define TILE_H 16
#define SW (TILE_W + 2)   // 258 halo'd tile width
#define SH (TILE_H + 2)   // 18  halo'd tile height
#define SSTRIDE 260       // padded LDS row stride (floats); 260*4 = 1040 B, 16B-aligned rows

typedef __attribute__((ext_vector_type(4))) float v4f;   // native vector: OK for nontemporal builtin

__device__ __forceinline__ void async_or_zero(const float* img, int gx, int gy,
                                              unsigned lds_byte, float* tile_base,
                                              unsigned lds_idx) {
    if ((unsigned)gy < (unsigned)IMG_H && (unsigned)gx < (unsigned)IMG_W) {
        const float* gptr = img + (size_t)gy * IMG_W + gx;
        // CDNA5 async global->LDS (ASYNCcnt-tracked, no VGPR round trip).
        // 'tile' is the only LDS object -> group-segment offset 0, so the
        // LDS address VGPR is just the byte offset into the tile.
        asm volatile("global_load_async_to_lds_b32 %0, %1, off"
                     :: "v"(lds_byte), "v"(gptr)
                     : "memory");
    } else {
        tile_base[lds_idx] = 0.0f;   // zero padding of the SAME conv
    }
}

__global__ __launch_bounds__(256) void sobel_tile_kernel(const float* __restrict__ x,
                                                         float* __restrict__ out) {
    __shared__ float tile[SH * SSTRIDE];

    const int tid = threadIdx.x;
    const int tx0 = blockIdx.x * TILE_W;
    const int ty0 = blockIdx.y * TILE_H;
    const int b   = blockIdx.z;

    const float* img = x + (size_t)b * (IMG_W * IMG_H);

    // ---- Halo'd tile load, division-free ----
    // Main body: col cc = tid covers lds cols [0,256) for all 18 rows.
#pragma unroll
    for (int r = 0; r < SH; ++r) {
        const int cc = tid;
        async_or_zero(img, tx0 - 1 + cc, ty0 - 1 + r,
                      (unsigned)(r * SSTRIDE + cc) * 4u, tile,
                      (unsigned)(r * SSTRIDE + cc));
    }
    // Tail: lds cols 256,257 for all 18 rows (36 elements, threads 0..35).
    if (tid < 2 * SH) {
        const int r  = tid >> 1;
        const int cc = 256 + (tid & 1);
        async_or_zero(img, tx0 - 1 + cc, ty0 - 1 + r,
                      (unsigned)(r * SSTRIDE + cc) * 4u, tile,
                      (unsigned)(r * SSTRIDE + cc));
    }

#if __has_builtin(__builtin_amdgcn_s_wait_asynccnt)
    __builtin_amdgcn_s_wait_asynccnt(0);
#else
    asm volatile("s_wait_asynccnt 0x0" ::: "memory");
#endif
    __syncthreads();

    // ---- Compute: each thread -> 4 rows x 4 cols (float4 per row) ----
    const int txg = (tid & 63) * 4;  // x within tile, multiple of 4 (aligned stores)
    const int ry  = tid >> 6;        // 0..3

    float* outimg = out + (size_t)b * (IMG_W * IMG_H);

#pragma unroll
    for (int k = 0; k < 4; ++k) {
        const int y = ry * 4 + k;    // output row within tile, 0..15

        float t[3][6];
#pragma unroll
        for (int r = 0; r < 3; ++r) {
            const float* row = &tile[(y + r) * SSTRIDE + txg];   // 16B-aligned
            const float4 v4 = *(const float4*)row;               // ds_load_b128
            const float2 v2 = *(const float2*)(row + 4);         // ds_load_b64
            t[r][0] = v4.x; t[r][1] = v4.y; t[r][2] = v4.z; t[r][3] = v4.w;
            t[r][4] = v2.x; t[r][5] = v2.y;
        }

        v4f res;
#pragma unroll
        for (int j = 0; j < 4; ++j) {
            // gx = (t02-t00) + 2*(t12-t10) + (t22-t20)   (FMA-folded)
            const float gxv = fmaf(2.0f, t[1][j + 2] - t[1][j],
                                   (t[0][j + 2] - t[0][j]) + (t[2][j + 2] - t[2][j]));
            // gy = (t20 + 2*t21 + t22) - (t00 + 2*t01 + t02)
            const float gyv = fmaf(2.0f, t[2][j + 1], t[2][j] + t[2][j + 2]) -
                              fmaf(2.0f, t[0][j + 1], t[0][j] + t[0][j + 2]);
            res[j] = __builtin_amdgcn_sqrtf(fmaf(gxv, gxv, gyv * gyv));
        }

        // NT store: output is write-once -> don't evict the (L2-resident) input.
        __builtin_nontemporal_store(
            res, (v4f*)&outimg[(size_t)(ty0 + y) * IMG_W + (tx0 + txg)]);
    }
}

extern "C" void kernel_launch(void* const* d_in, const int* in_sizes, int n_in,
                              void* d_out, int out_size, void* d_ws, size_t ws_size,
                              hipStream_t stream) {
    const float* x = (const float*)d_in[0];
    float* out     = (float*)d_out;

    const int batch = in_sizes[0] / (IMG_W * IMG_H);   // 32

    dim3 grid(IMG_W / TILE_W, IMG_H / TILE_H, batch);  // (4, 64, 32)
    sobel_tile_kernel<<<grid, 256, 0, stream>>>(x, out);
}